// LTGNN_Encoder_35003983462573
// MI455X (gfx1250) — compile-verified
//
#include <hip/hip_runtime.h>
#include <hip/hip_bf16.h>
#include <stdint.h>

// CDNA5 / gfx1250. Wave32. D (emb size) = 64 floats = 256B per row.

typedef __attribute__((ext_vector_type(2))) float v2f;
typedef __attribute__((ext_vector_type(8))) float v8f;

#define AS1 __attribute__((address_space(1)))
#define AS3 __attribute__((address_space(3)))

#define BETA  0.9f
#define ALPHA 0.1f

#if defined(__has_builtin)
#  if __has_builtin(__builtin_amdgcn_global_load_async_to_lds_b32)
#    define HAVE_ASYNC_LDS 1
#  endif
#  if __has_builtin(__builtin_amdgcn_s_wait_asynccnt)
#    define HAVE_WAIT_ASYNC 1
#  endif
#endif
#ifndef HAVE_ASYNC_LDS
#  define HAVE_ASYNC_LDS 0
#endif
#ifndef HAVE_WAIT_ASYNC
#  define HAVE_WAIT_ASYNC 0
#endif

// ---------------------------------------------------------------------------
// Pass 1: y = BETA * (e_out - in_mem)   (float4 vectorized, one DRAM pass)
// ---------------------------------------------------------------------------
__global__ void k_prep(const float4* __restrict__ e_out,
                       const float4* __restrict__ in_mem,
                       float4* __restrict__ y, int n4) {
  int i = blockIdx.x * blockDim.x + threadIdx.x;
  if (i < n4) {
    float4 a = e_out[i];
    float4 b = in_mem[i];
    float4 r;
    r.x = BETA * (a.x - b.x);
    r.y = BETA * (a.y - b.y);
    r.z = BETA * (a.z - b.z);
    r.w = BETA * (a.w - b.w);
    y[i] = r;
  }
}

// ---------------------------------------------------------------------------
// Pass 2: out = BETA*aggr + ALPHA*e_in via V_WMMA_F32_16X16X4_F32
//   D = (BETA * I16) x G + C,  C preloaded with ALPHA*H in the C/D layout.
//   Per 16x16 tile: 4 chained WMMAs over K-chunks of 4.
// Layouts (ISA 7.12.2, 32-bit, wave32):
//   A 16x4 : lane m = l&15 (both halves), K = vgpr + 2*(l>>4)
//   B 4x16 : lane n = l&15,               K = vgpr + 2*(l>>4)
//   C/D    : lane n = l&15,               M = vgpr + 8*(l>>4)
// ---------------------------------------------------------------------------
__global__ void k_init_wmma(const float* __restrict__ aggr,
                            const float* __restrict__ e_in,
                            float* __restrict__ out, int ntiles) {
  const int wave  = threadIdx.x >> 5;                 // 8 waves / block
  const int gtile = blockIdx.x * 8 + wave;            // wave-uniform
  if (gtile >= ntiles) return;                        // whole-wave exit: EXEC stays all-1s
  const int tx = gtile & 3;                           // 4 tiles across D=64
  const int ty = gtile >> 2;
  const int row0 = ty * 16;
  const int col0 = tx * 16;

  const int lane = threadIdx.x & 31;
  const int hi   = lane >> 4;
  const int m    = lane & 15;                         // A-row / B-col / C-col index

  // C = ALPHA * e_in tile, in C/D layout
  v8f c;
#pragma unroll
  for (int r = 0; r < 8; ++r)
    c[r] = ALPHA * e_in[(size_t)(row0 + r + 8 * hi) * 64 + col0 + m];

  // Accumulate (BETA*I) x aggr over 4 K-chunks
#pragma unroll
  for (int q = 0; q < 4; ++q) {
    const int k0 = 2 * hi;                            // K of vgpr slot 0
    v2f a, b;
    a.x = (m == q * 4 + k0)     ? BETA : 0.0f;        // A = BETA * I16 columns [4q..4q+3]
    a.y = (m == q * 4 + k0 + 1) ? BETA : 0.0f;
    const float* g = aggr + (size_t)(row0 + q * 4 + k0) * 64 + col0 + m;
    b.x = g[0];                                       // B = aggr rows [4q..4q+3]
    b.y = g[64];
    c = __builtin_amdgcn_wmma_f32_16x16x4_f32(
        false, a, false, b, (short)0, c, false, false);
  }

#pragma unroll
  for (int r = 0; r < 8; ++r)
    out[(size_t)(row0 + r + 8 * hi) * 64 + col0 + m] = c[r];
}

// Tail rows (N % 16 != 0) handled elementwise.
__global__ void k_init_tail(const float* __restrict__ aggr,
                            const float* __restrict__ e_in,
                            float* __restrict__ out, int offset, int n) {
  int i = blockIdx.x * blockDim.x + threadIdx.x;
  if (i < n) {
    int idx = offset + i;
    out[idx] = BETA * aggr[idx] + ALPHA * e_in[idx];
  }
}

// ---------------------------------------------------------------------------
// Pass 3: edge scatter. 64 edges per block; edge triples staged into LDS via
// async load-to-LDS (gfx1250 path), then 16 lanes per edge gather y[col] as
// float4 and do 4 no-return f32 atomics into out[row]. All gather/scatter
// traffic stays in the 192MB L2 (y is 25.6MB).
// If use_y == 0 (workspace too small), gather e_out/in_mem directly instead.
// ---------------------------------------------------------------------------
__global__ void k_edges(const int* __restrict__ rows,
                        const int* __restrict__ cols,
                        const float* __restrict__ vals,
                        const float* __restrict__ y,
                        const float* __restrict__ e_out,
                        const float* __restrict__ in_mem,
                        float* __restrict__ out, int nE, int use_y) {
  __shared__ int   s_row[64];
  __shared__ int   s_col[64];
  __shared__ float s_val[64];

  const int t    = threadIdx.x;
  const int base = blockIdx.x * 64;

  if (t < 64 && (base + t) < nE) {
#if HAVE_ASYNC_LDS
    // Builtin signature (per hipcc diagnostic): (__device__ int*, __shared__ int*, imm, imm)
    __builtin_amdgcn_global_load_async_to_lds_b32(
        (AS1 int*)(AS1 void*)(rows + base + t),
        (AS3 int*)(AS3 void*)&s_row[t], 0, 0);
    __builtin_amdgcn_global_load_async_to_lds_b32(
        (AS1 int*)(AS1 void*)(cols + base + t),
        (AS3 int*)(AS3 void*)&s_col[t], 0, 0);
    __builtin_amdgcn_global_load_async_to_lds_b32(
        (AS1 int*)(AS1 void*)(vals + base + t),
        (AS3 int*)(AS3 void*)&s_val[t], 0, 0);
#else
    s_row[t] = rows[base + t];
    s_col[t] = cols[base + t];
    s_val[t] = vals[base + t];
#endif
  }
#if HAVE_ASYNC_LDS
#  if HAVE_WAIT_ASYNC
  __builtin_amdgcn_s_wait_asynccnt(0);
#  else
  asm volatile("s_wait_asynccnt 0" ::: "memory");
#  endif
#endif
  __syncthreads();

  const int sub = t >> 4;   // edge slot within a pass (16 edges / 256 threads)
  const int l16 = t & 15;   // float4 chunk within the 64-float row
#pragma unroll
  for (int p = 0; p < 4; ++p) {
    const int el = p * 16 + sub;
    const int e  = base + el;
    if (e < nE) {
      const int   r = s_row[el];
      const int   cdx = s_col[el];
      const float v = s_val[el];
      float4 q;
      if (use_y) {
        q = ((const float4*)(y + (size_t)cdx * 64))[l16];      // y already has BETA folded in
      } else {
        float4 a = ((const float4*)(e_out + (size_t)cdx * 64))[l16];
        float4 b = ((const float4*)(in_mem + (size_t)cdx * 64))[l16];
        q.x = BETA * (a.x - b.x); q.y = BETA * (a.y - b.y);
        q.z = BETA * (a.z - b.z); q.w = BETA * (a.w - b.w);
      }
      float* dst = out + (size_t)r * 64 + (size_t)l16 * 4;
      atomicAdd(dst + 0, v * q.x);   // result unused -> no-return global_atomic_add_f32
      atomicAdd(dst + 1, v * q.y);
      atomicAdd(dst + 2, v * q.z);
      atomicAdd(dst + 3, v * q.w);
    }
  }
}

// ---------------------------------------------------------------------------
extern "C" void kernel_launch(void* const* d_in, const int* in_sizes, int n_in,
                              void* d_out, int out_size, void* d_ws, size_t ws_size,
                              hipStream_t stream) {
  const int*   rows    = (const int*)d_in[0];
  const int*   cols    = (const int*)d_in[1];
  const float* vals    = (const float*)d_in[2];
  const float* e_in    = (const float*)d_in[3];
  const float* e_out   = (const float*)d_in[4];
  const float* in_mem  = (const float*)d_in[5];
  const float* in_aggr = (const float*)d_in[6];
  float* out = (float*)d_out;
  float* y   = (float*)d_ws;

  const int nE = in_sizes[0];
  const int ND = in_sizes[3];       // N * 64
  const int N  = ND / 64;

  const int use_y = (ws_size >= (size_t)ND * sizeof(float)) ? 1 : 0;

  // Pass 1: y = BETA*(e_out - in_mem)
  if (use_y) {
    const int n4 = ND / 4;
    k_prep<<<(n4 + 255) / 256, 256, 0, stream>>>(
        (const float4*)e_out, (const float4*)in_mem, (float4*)y, n4);
  }

  // Pass 2: out = BETA*in_aggr + ALPHA*e_in  (WMMA tiles + scalar tail)
  const int nty    = N / 16;
  const int ntiles = nty * 4;
  if (ntiles > 0)
    k_init_wmma<<<(ntiles + 7) / 8, 256, 0, stream>>>(in_aggr, e_in, out, ntiles);
  const int tailRows = N - nty * 16;
  if (tailRows > 0) {
    const int tail = tailRows * 64;
    k_init_tail<<<(tail + 255) / 256, 256, 0, stream>>>(in_aggr, e_in, out,
                                                        nty * 16 * 64, tail);
  }

  // Pass 3: scatter-add edge contributions
  if (nE > 0)
    k_edges<<<(nE + 63) / 64, 256, 0, stream>>>(rows, cols, vals, y, e_out,
                                                in_mem, out, nE, use_y);
}